// LocalConcatSheafLearner_55628416418071
// MI455X (gfx1250) — compile-verified
//
#include <hip/hip_runtime.h>

typedef __attribute__((ext_vector_type(2))) float v2f;
typedef __attribute__((ext_vector_type(8))) float v8f;

#define D_FEAT   128
#define TWO_D    256
#define N_OUT    16
#define K_STEPS  64                   // 256 / 4
#define LDS_B_FLOATS (K_STEPS * 64)   // 64 steps * 64 floats = 4096 floats = 16 KB

__device__ __forceinline__ float fast_tanh(float v) {
#if defined(__has_builtin)
#if __has_builtin(__builtin_amdgcn_tanhf)
    return __builtin_amdgcn_tanhf(v);   // hw V_TANH_F32 (gfx1250 trans op)
#else
    return tanhf(v);
#endif
#else
    return tanhf(v);
#endif
}

// D = tanh(A * B), A = gathered+concat features [16 x 256], B = W^T [256 x 16].
// One wave32 handles one tile of 16 edges using V_WMMA_F32_16X16X4_F32.
__launch_bounds__(256, 2)
__global__ void sheaf_wmma_kernel(const float* __restrict__ x,
                                  const long long* __restrict__ ei,   // [2, E]
                                  const float* __restrict__ W,        // [16, 256] row-major
                                  float* __restrict__ out,            // [E, 16]
                                  int E, int numTiles)
{
    // ---- Stage W into LDS, pre-permuted into WMMA B-operand lane order ----
    // For step j (K = 4j..4j+3), lane l wants float2:
    //   { B[2h+0][n], B[2h+1][n] } with n = l&15, h = l>>4, B[k'][n] = W[n][4j+k'].
    // Stored contiguously: ldsB[j*64 + l*2 + c] = W[n*256 + 4j + 2h + c].
    __shared__ float ldsB[LDS_B_FLOATS];
    for (int f = threadIdx.x; f < LDS_B_FLOATS; f += blockDim.x) {
        int j = f >> 6;
        int r = f & 63;
        int l = r >> 1;
        int c = r & 1;
        int n = l & 15;
        int h = l >> 4;
        ldsB[f] = W[n * TWO_D + 4 * j + 2 * h + c];
    }
    __syncthreads();

    const int lane          = threadIdx.x & 31;
    const int waveId        = threadIdx.x >> 5;
    const int wavesPerBlock = blockDim.x >> 5;
    const int globalWave    = blockIdx.x * wavesPerBlock + waveId;
    const int totalWaves    = gridDim.x * wavesPerBlock;

    const int m = lane & 15;   // edge-in-tile (A-matrix row M)
    const int h = lane >> 4;   // K half-select per the 16x4 f32 A layout

    const v2f* lb = (const v2f*)ldsB;

    for (int tile = globalWave; tile < numTiles; tile += totalWaves) {
        int e  = tile * 16 + m;
        int ec = (e < E) ? e : (E - 1);          // clamp (EXEC must stay all-1 for WMMA)
        long long rowi = ei[ec];
        long long coli = ei[(long long)E + ec];
        const float* __restrict__ pr = x + rowi * (long long)D_FEAT;
        const float* __restrict__ pc = x + coli * (long long)D_FEAT;

        v8f acc = {};

        // K = 0..127  -> x[row]
        #pragma unroll 4
        for (int j = 0; j < 32; ++j) {
            v2f a = *(const v2f*)(pr + 4 * j + 2 * h);     // A: global_load_b64 gather
            v2f b = lb[j * 32 + lane];                     // B: ds_load_b64, conflict-free
            acc = __builtin_amdgcn_wmma_f32_16x16x4_f32(
                      false, a, false, b, (short)0, acc, false, false);
        }
        // K = 128..255 -> x[col]
        #pragma unroll 4
        for (int j = 0; j < 32; ++j) {
            v2f a = *(const v2f*)(pc + 4 * j + 2 * h);
            v2f b = lb[(32 + j) * 32 + lane];
            acc = __builtin_amdgcn_wmma_f32_16x16x4_f32(
                      false, a, false, b, (short)0, acc, false, false);
        }

        // Epilogue: tanh + store. C/D layout: VGPR r, lanes 0-15 -> M=r, N=lane;
        // lanes 16-31 -> M=r+8, N=lane-16.
        // Lane's elements live at out[(base + 8h + r)*16 + m], r = 0..7, i.e.
        // one base pointer + immediate offsets r*64 bytes.
        long long base = (long long)tile * 16;
        if (base + 16 <= (long long)E) {
            // Full tile (the common case: E % 16 == 0): unguarded clause of stores.
            float* __restrict__ po = out + (base + 8 * h) * N_OUT + m;
            #pragma unroll
            for (int r = 0; r < 8; ++r) {
                po[r * N_OUT] = fast_tanh(acc[r]);
            }
        } else {
            #pragma unroll
            for (int r = 0; r < 8; ++r) {
                long long eo = base + r + 8 * h;
                if (eo < E) out[eo * N_OUT + m] = fast_tanh(acc[r]);
            }
        }
    }
}

extern "C" void kernel_launch(void* const* d_in, const int* in_sizes, int n_in,
                              void* d_out, int out_size, void* d_ws, size_t ws_size,
                              hipStream_t stream) {
    const float*     x  = (const float*)d_in[0];       // [100000, 128] f32
    const long long* ei = (const long long*)d_in[1];   // [2, 1600000] i64
    const float*     W  = (const float*)d_in[2];       // [16, 256] f32
    float*           out = (float*)d_out;              // [E, 16] f32

    const int E        = in_sizes[1] / 2;
    const int numTiles = (E + 15) / 16;

    const int threads = 256;                 // 8 waves/block
    int blocks = 1280;                       // ~10240 waves, grid-stride over 100K tiles
    int maxBlocks = (numTiles + 7) / 8;
    if (blocks > maxBlocks) blocks = maxBlocks;
    if (blocks < 1) blocks = 1;

    sheaf_wmma_kernel<<<blocks, threads, 0, stream>>>(x, ei, W, out, E, numTiles);
}